// IoU_21698174779527
// MI455X (gfx1250) — compile-verified
//
#include <hip/hip_runtime.h>

typedef __attribute__((ext_vector_type(2))) float v2f;
typedef __attribute__((ext_vector_type(8))) float v8f;

#define NCLS 19
#define CPAD 32   // padded bin count (LDS + ws layout), zeroed so extra bins are inert

// ---------------------------------------------------------------- zero scratch
__global__ void iou_zero_ws(unsigned* __restrict__ w) {
    int t = blockIdx.x * blockDim.x + threadIdx.x;
    if (t < 3 * CPAD) w[t] = 0u;
}

// ---------------------------------------------------------------- bulk phase
// Each thread handles 4 consecutive pixels (float4/int4 loads). Channel stride
// is the HW plane, so lane-consecutive float4 loads are fully coalesced per
// channel -> pure HBM stream (~336 MB, ~14 us floor at 23.3 TB/s).
__global__ __launch_bounds__(256) void iou_hist(const float*  __restrict__ preds,
                                                const int*    __restrict__ targets,
                                                unsigned*     __restrict__ cnt,
                                                int nGroups, int plane)
{
    __shared__ unsigned s_inter[CPAD], s_pred[CPAD], s_true[CPAD];
    const int tid = threadIdx.x;
    if (tid < CPAD) { s_inter[tid] = 0u; s_pred[tid] = 0u; s_true[tid] = 0u; }
    __syncthreads();

    const int str4 = plane >> 2;                 // channel stride in float4 units
    const int step = gridDim.x * blockDim.x;     // grid stride in groups

    for (int g = blockIdx.x * blockDim.x + tid; g < nGroups; g += step) {
        const int b  = g / str4;                 // batch index
        const int sg = g - b * str4;             // float4 index within plane
        const float4* p = (const float4*)preds + (size_t)b * NCLS * str4 + sg;

        // gfx1250 prefetch of (approximately) the next grid-stride group's line;
        // provably in-bounds for this problem size.
        __builtin_prefetch((const void*)(p + step), 0, 1);

        float4 best = p[0];
        int bx = 0, by = 0, bz = 0, bw = 0;
        #pragma unroll
        for (int c = 1; c < NCLS; ++c) {
            float4 v = p[(size_t)c * str4];
            if (v.x > best.x) { best.x = v.x; bx = c; }   // first-max == jnp.argmax
            if (v.y > best.y) { best.y = v.y; by = c; }
            if (v.z > best.z) { best.z = v.z; bz = c; }
            if (v.w > best.w) { best.w = v.w; bw = c; }
        }

        const int4 t = ((const int4*)targets)[g];

        atomicAdd(&s_pred[bx], 1u);  atomicAdd(&s_pred[by], 1u);
        atomicAdd(&s_pred[bz], 1u);  atomicAdd(&s_pred[bw], 1u);
        atomicAdd(&s_true[t.x], 1u); atomicAdd(&s_true[t.y], 1u);
        atomicAdd(&s_true[t.z], 1u); atomicAdd(&s_true[t.w], 1u);
        if (bx == t.x) atomicAdd(&s_inter[bx], 1u);
        if (by == t.y) atomicAdd(&s_inter[by], 1u);
        if (bz == t.z) atomicAdd(&s_inter[bz], 1u);
        if (bw == t.w) atomicAdd(&s_inter[bw], 1u);
    }
    __syncthreads();

    if (tid < CPAD) {
        unsigned vi = s_inter[tid], vp = s_pred[tid], vt = s_true[tid];
        if (vi) atomicAdd(&cnt[tid],            vi);
        if (vp) atomicAdd(&cnt[CPAD + tid],     vp);
        if (vt) atomicAdd(&cnt[2 * CPAD + tid], vt);
    }
}

// ---------------------------------------------------------------- epilogue
// One wave32. Per-class IoU per lane, then the 18-element mean-sum is done with
// V_WMMA_F32_16X16X4_F32: A holds the values (16x4 f32, 2 VGPRs/lane per ISA
// layout), B = all-ones (layout independent), so D[M][N] = rowsum(A,M).
// Lane 0 / lane 16 accumulators hold column-0 row sums; shfl_xor(16) finishes.
__global__ void iou_finalize(const unsigned* __restrict__ cnt,
                             float* __restrict__ out)
{
    const int l = threadIdx.x;                       // 0..31, full wave
    const unsigned vi = cnt[l];
    const unsigned vp = cnt[CPAD + l];
    const unsigned vt = cnt[2 * CPAD + l];
    const unsigned un = vp + vt - vi;
    const float iou = (vt > 0u) ? ((float)vi / (float)(un > 1u ? un : 1u)) : 0.0f;

    if (l >= 1 && l < NCLS) out[l - 1] = iou;        // iou = iou_all[1:]

    const float val = (l >= 1 && l < NCLS) ? iou : 0.0f;
    v2f a; a[0] = val;  a[1] = 0.0f;                 // A[M][K]: K=0 (lanes 0-15), K=2 (16-31)
    v2f b; b[0] = 1.0f; b[1] = 1.0f;                 // B = ones
    v8f acc = {};
    acc = __builtin_amdgcn_wmma_f32_16x16x4_f32(
        /*neg_a=*/false, a, /*neg_b=*/false, b,
        /*c_mod=*/(short)0, acc, /*reuse_a=*/false, /*reuse_b=*/false);

    float part = acc[0] + acc[1] + acc[2] + acc[3] + acc[4] + acc[5] + acc[6] + acc[7];
    part += __shfl_xor(part, 16, 32);                // rows 0-7 (+) rows 8-15 @ N=0
    if (l == 0) out[NCLS - 1] = part * (1.0f / 18.0f);
}

// ---------------------------------------------------------------- launcher
extern "C" void kernel_launch(void* const* d_in, const int* in_sizes, int n_in,
                              void* d_out, int out_size, void* d_ws, size_t ws_size,
                              hipStream_t stream) {
    const float* preds   = (const float*)d_in[0];
    const int*   targets = (const int*)d_in[1];
    unsigned*    cnt     = (unsigned*)d_ws;
    float*       out     = (float*)d_out;

    const int plane   = 512 * 1024;          // H*W
    const int nGroups = in_sizes[1] / 4;     // B*H*W / 4 pixels per thread-group

    iou_zero_ws<<<1, 128, 0, stream>>>(cnt);
    iou_hist<<<1024, 256, 0, stream>>>(preds, targets, cnt, nGroups, plane);
    iou_finalize<<<1, 32, 0, stream>>>(cnt, out);
}